// GeneralizedInteractionFusion_35450660061571
// MI455X (gfx1250) — compile-verified
//
#include <hip/hip_runtime.h>

// out[b,n,D] = sum_d ( B0[b]^T · alpha[:,:,n] · Bi[b] )[D,d] * W[n,D,d] * h[n,d]
// B0: (256,40,64)  Bi: (256,40,64)  W: (40,64,64)  alpha: (40,40,40)  h: (40,64,1)
// out: (256,40,64) fp32

typedef float v2f __attribute__((ext_vector_type(2)));
typedef float v8f __attribute__((ext_vector_type(8)));

#define NF     40   // fields (f, i, n)
#define ED     64   // embed dim (D, d)
#define FPAD   48   // f/i padded to multiple of 16
// LDS row strides (padded to avoid bank conflicts on strided A-fragment reads)
#define SB0_S  49   // sB0T[D][f]   (stride 49: gcd(49,64)=1)
#define SBI_S  65   // sBi [i][d]
#define SC1_S  65   // sC1 [f][d]
#define SAL_S  49   // sAl [f][i]
#define SWH_S  68   // sWh [D][d]   (half-wave rows land on disjoint banks)

__global__ __launch_bounds__(256)
void gif_wmma_kernel(const float* __restrict__ B0, const float* __restrict__ Bi,
                     const float* __restrict__ W,  const float* __restrict__ alpha,
                     const float* __restrict__ h,  float* __restrict__ out)
{
    __shared__ float sB0T[ED   * SB0_S];   // B0[b] transposed: [D][f], f padded w/ zeros
    __shared__ float sBi [FPAD * SBI_S];   // Bi[b]: [i][d], i padded w/ zeros
    __shared__ float sC1 [FPAD * SC1_S];   // GEMM1 result: [f][d]
    __shared__ float sAl [FPAD * SAL_S];   // alpha[:,:,n]: [f][i], padded w/ zeros
    __shared__ float sWh [ED   * SWH_S];   // W[n]*h[n]: [D][d]
    __shared__ float sOut[ED];

    const int b    = blockIdx.x;
    const int tid  = threadIdx.x;
    const int wid  = tid >> 5;             // 8 waves of 32
    const int lane = tid & 31;
    const int l16  = lane & 15;
    const int hv   = lane >> 4;            // half-wave: 0 or 1
    const int kh   = hv << 1;              // K sub-offset within a k=4 step: 0 or 2

    // Zero padded regions once (pad stays zero; live regions rewritten each use).
    for (int i = tid; i < ED   * SB0_S; i += 256) sB0T[i] = 0.f;
    for (int i = tid; i < FPAD * SBI_S; i += 256) sBi[i]  = 0.f;
    for (int i = tid; i < FPAD * SAL_S; i += 256) sAl[i]  = 0.f;
    __syncthreads();

    // Stage per-batch operands: B0[b] (transposed) and Bi[b].
    const float* B0b = B0 + (size_t)b * NF * ED;
    const float* Bib = Bi + (size_t)b * NF * ED;
    for (int idx = tid; idx < NF * ED; idx += 256) {
        int f = idx >> 6, d = idx & 63;
        sB0T[d * SB0_S + f] = B0b[idx];
        sBi [f * SBI_S + d] = Bib[idx];
    }
    __syncthreads();

    for (int n = 0; n < NF; ++n) {
        // alpha[:,:,n] gather (innermost-stride-40) -> sAl[f][i]
        for (int idx = tid; idx < NF * NF; idx += 256)
            sAl[(idx / NF) * SAL_S + (idx % NF)] = alpha[(size_t)idx * NF + n];

        // sWh[D][d] = W[n,D,d] * h[n,d]
        const float* Wn = W + (size_t)n * ED * ED;
        const float* hn = h + (size_t)n * ED;
        for (int idx = tid; idx < ED * ED; idx += 256) {
            int D = idx >> 6, d = idx & 63;
            sWh[D * SWH_S + d] = Wn[idx] * hn[d];
        }
        if (tid < ED) sOut[tid] = 0.f;

        // Warm L2/WGP$ for next iteration's W (gfx1250 global_prefetch_b8).
        if (n + 1 < NF && tid == 0) {
            __builtin_prefetch(Wn + (size_t)ED * ED, 0, 3);
            __builtin_prefetch(hn + ED, 0, 3);
        }
        __syncthreads();

        // ---- GEMM1: C1(48x64) = alpha_n(48x48) x Bi(48x64), K = 48 ----
        for (int t = wid; t < 12; t += 8) {
            const int Mt = t >> 2, Nt = t & 3;
            const int m  = Mt * 16 + l16;      // row (f)
            const int nc = Nt * 16 + l16;      // col (d)
            v8f acc = {};
#pragma unroll
            for (int kk = 0; kk < 12; ++kk) {
                const int kb = kk * 4 + kh;
                v2f a, bf;
                a.x  = sAl[m * SAL_S + kb];
                a.y  = sAl[m * SAL_S + kb + 1];
                bf.x = sBi[kb       * SBI_S + nc];
                bf.y = sBi[(kb + 1) * SBI_S + nc];
                acc = __builtin_amdgcn_wmma_f32_16x16x4_f32(
                          false, a, false, bf, (short)0, acc, false, false);
            }
#pragma unroll
            for (int v = 0; v < 8; ++v) {
                const int row = Mt * 16 + v + (hv << 3);
                sC1[row * SC1_S + nc] = acc[v];
            }
        }
        __syncthreads();

        // ---- GEMM2: fusion(64x64) = B0^T(64x48) x C1(48x64), fused epilogue ----
        for (int t = wid; t < 16; t += 8) {
            const int Mt = t >> 2, Nt = t & 3;
            const int m  = Mt * 16 + l16;      // row (D)
            const int nc = Nt * 16 + l16;      // col (d)
            v8f acc = {};
#pragma unroll
            for (int kk = 0; kk < 12; ++kk) {
                const int kb = kk * 4 + kh;
                v2f a, bf;
                a.x  = sB0T[m * SB0_S + kb];
                a.y  = sB0T[m * SB0_S + kb + 1];
                bf.x = sC1[kb       * SC1_S + nc];
                bf.y = sC1[(kb + 1) * SC1_S + nc];
                acc = __builtin_amdgcn_wmma_f32_16x16x4_f32(
                          false, a, false, bf, (short)0, acc, false, false);
            }
            // out[D] += sum over this tile's 16 d-columns of fusion*Wh
#pragma unroll
            for (int v = 0; v < 8; ++v) {
                const int row = Mt * 16 + v + (hv << 3);   // D in 0..63
                float p = acc[v] * sWh[row * SWH_S + nc];
                p += __shfl_xor(p, 1, 16);                 // butterfly over the 16
                p += __shfl_xor(p, 2, 16);                 // lanes holding one row
                p += __shfl_xor(p, 4, 16);
                p += __shfl_xor(p, 8, 16);
                if (l16 == 0) atomicAdd(&sOut[row], p);    // ds_add_f32
            }
        }
        __syncthreads();

        if (tid < ED)
            out[((size_t)b * NF + n) * ED + tid] = sOut[tid];
        __syncthreads();   // sAl/sWh/sOut reused next iteration
    }
}

extern "C" void kernel_launch(void* const* d_in, const int* in_sizes, int n_in,
                              void* d_out, int out_size, void* d_ws, size_t ws_size,
                              hipStream_t stream) {
    const float* B0    = (const float*)d_in[0];
    const float* Bi    = (const float*)d_in[1];
    const float* W     = (const float*)d_in[2];
    const float* alpha = (const float*)d_in[3];
    const float* h     = (const float*)d_in[4];
    float* out = (float*)d_out;
    (void)in_sizes; (void)n_in; (void)out_size; (void)d_ws; (void)ws_size;

    gif_wmma_kernel<<<dim3(256), dim3(256), 0, stream>>>(B0, Bi, W, alpha, h, out);
}